// RotMat_43250320670841
// MI455X (gfx1250) — compile-verified
//
#include <hip/hip_runtime.h>

// ---------------------------------------------------------------------------
// U = (G_K ... G_1) X  with N=M=512, D=4096, K = 512*511/2 = 130816 rotations.
//
//   k0: cs/sn = cos/sin(thetas)
//   k1: build R = G_K...G_1 * I (512x512), wavefront schedule: rotation (i,j)
//       runs at stage t = i+j-1 -> 1021 stages, up to 256 independent
//       rotations/stage (row sets provably disjoint). 32-column LDS slice/WG.
//   k2: U = R * X as fp32 WMMA GEMM (v_wmma_f32_16x16x4_f32).
//       Block = 8 waves = 128 rows x 64 cols. X k-slabs (32x64) staged in LDS
//       K-major so every B fragment is one ds_load_b64 straight into the WMMA
//       operand. Per-slab A fragments are loaded BEFORE the next-slab X
//       prefetch so compiler loadcnt waits never retire the prefetch early.
// ---------------------------------------------------------------------------

typedef __attribute__((ext_vector_type(2))) float v2f;
typedef __attribute__((ext_vector_type(4))) float v4f;
typedef __attribute__((ext_vector_type(8))) float v8f;

#define N_ROT 130816
#define NSZ   512
#define DSZ   4096
#define C_TILE 32           // R-columns per WG in the rotation kernel
#define KSLAB 32            // k-rows of X staged in LDS per iteration
#define KS_STRIDE 34        // K-major LDS column stride (even -> b64 aligned)

// ---------------- kernel 0: cos/sin of all thetas --------------------------
__global__ __launch_bounds__(256) void rot_cossin(const float* __restrict__ th,
                                                  float* __restrict__ cs,
                                                  float* __restrict__ sn) {
    int k = blockIdx.x * 256 + threadIdx.x;
    if (k < N_ROT) {
        float t = th[k];
        cs[k] = __builtin_cosf(t);
        sn[k] = __builtin_sinf(t);
    }
}

// ---------------- kernel 1: wavefront-scheduled build of R -----------------
__global__ __launch_bounds__(256) void rot_build_R(const float* __restrict__ cs,
                                                   const float* __restrict__ sn,
                                                   float* __restrict__ Rm) {
    extern __shared__ float lds[];          // [512][C_TILE]
    const int tid  = threadIdx.x;
    const int col0 = blockIdx.x * C_TILE;

    for (int idx = tid; idx < NSZ * C_TILE; idx += 256) {
        int row = idx >> 5, c = idx & 31;
        lds[idx] = (row == col0 + c) ? 1.0f : 0.0f;
    }

    for (int t = 0; t <= 2 * (NSZ - 1) - 2; ++t) {   // t = 0..1020
        __syncthreads();
        int ilo = t - (NSZ - 2); if (ilo < 0) ilo = 0;   // j = t+1-i <= 511
        int ihi = t >> 1;                                 // j > i
        int nrot = ihi - ilo + 1;
        // one wave == one rotation across the 32-column slice:
        // conflict-free LDS, wave-uniform cs/sn loads.
        for (int idx = tid; idx < nrot * C_TILE; idx += 256) {
            int r = idx >> 5, c = idx & 31;
            int i = ilo + r;
            int j = t + 1 - i;
            int kk = i * (NSZ - 1) - ((i * (i - 1)) >> 1) + (j - i - 1);
            float cc = cs[kk], ss = sn[kk];
            float ui = lds[i * C_TILE + c];
            float uj = lds[j * C_TILE + c];
            lds[i * C_TILE + c] = cc * ui - ss * uj;
            lds[j * C_TILE + c] = ss * ui + cc * uj;
        }
    }
    __syncthreads();

    for (int idx = tid; idx < NSZ * C_TILE; idx += 256) {
        int row = idx >> 5, c = idx & 31;
        Rm[row * NSZ + col0 + c] = lds[idx];
    }
}

// ---------------- kernel 2: U = R * X via fp32 WMMA + LDS-staged B ---------
// ISA layouts (CDNA5 ISA):
//   A 16x4 f32 : lanes 0-15 -> M=lane, K={k,k+1}; lanes 16-31 -> K={k+2,k+3}
//   B 4x16 f32 : lanes 0-15 -> N=lane, K={k,k+1}; lanes 16-31 -> K={k+2,k+3}
//   C/D 16x16  : VGPR v -> row v (lanes 0-15) / row v+8 (lanes 16-31)
__global__ __launch_bounds__(256) void rot_wmma_gemm(const float* __restrict__ Rm,
                                                     const float* __restrict__ X,
                                                     float* __restrict__ U) {
    // X slab, K-major per column: xs[c * 34 + k], c in [0,64), k in [0,32).
    // All B-fragment dword indices are even -> natural b64 alignment.
    __shared__ float xs[64 * KS_STRIDE];

    const int tid  = threadIdx.x;
    const int lane = tid & 31;
    const int wid  = tid >> 5;            // 8 waves
    const int half = lane >> 4;           // which K pair this lane holds
    const int l16  = lane & 15;

    const int bm     = blockIdx.x >> 6;   // 4 groups of 8 row-tiles
    const int cstrip = blockIdx.x & 63;   // 64 col strips of 64
    const int m0   = (bm * 8 + wid) << 4; // this wave's 16-row tile
    const int col0 = cstrip << 6;

    v8f acc0 = {0,0,0,0,0,0,0,0};
    v8f acc1 = {0,0,0,0,0,0,0,0};
    v8f acc2 = {0,0,0,0,0,0,0,0};
    v8f acc3 = {0,0,0,0,0,0,0,0};

    const float* arow = Rm + (m0 + l16) * NSZ + 2 * half;  // A fragment base
    const float* xg   = X + col0;

    // slab-load indexing: 512 float4s per slab, 2 per thread
    const int i0 = tid, i1 = tid + 256;
    const int k0r = i0 >> 4, c0r = (i0 & 15) << 2;  // (k-row, col) of r0
    const int k1r = i1 >> 4, c1r = (i1 & 15) << 2;  // (k-row, col) of r1

    // prefetch slab 0
    v4f r0 = *(const v4f*)(xg + k0r * DSZ + c0r);
    v4f r1 = *(const v4f*)(xg + k1r * DSZ + c1r);

    // per-lane LDS read base: column l16, K pair offset 2*half
    const float* bcol = &xs[l16 * KS_STRIDE + 2 * half];

    for (int s = 0; s < NSZ / KSLAB; ++s) {          // 16 slabs
        // stage current slab into LDS, transposed to K-major
#pragma unroll
        for (int j = 0; j < 4; ++j) xs[(c0r + j) * KS_STRIDE + k0r] = r0[j];
#pragma unroll
        for (int j = 0; j < 4; ++j) xs[(c1r + j) * KS_STRIDE + k1r] = r1[j];
        __syncthreads();

        // A fragments for this slab: issued BEFORE the X prefetch, so waits
        // on them leave the prefetch outstanding (loads retire in order).
        v2f a[KSLAB / 4];
#pragma unroll
        for (int kk = 0; kk < KSLAB / 4; ++kk)
            a[kk] = *(const v2f*)(arow + s * KSLAB + kk * 4);

        // next slab's X loads: overlap the 32 WMMAs below
        if (s + 1 < NSZ / KSLAB) {
            const float* xn = xg + (s + 1) * KSLAB * DSZ;
            r0 = *(const v4f*)(xn + k0r * DSZ + c0r);
            r1 = *(const v4f*)(xn + k1r * DSZ + c1r);
        }

#pragma unroll
        for (int kk = 0; kk < KSLAB / 4; ++kk) {     // 8 k-steps of 4
            v2f b0 = *(const v2f*)(bcol + 0 * 16 * KS_STRIDE + kk * 4);
            v2f b1 = *(const v2f*)(bcol + 1 * 16 * KS_STRIDE + kk * 4);
            v2f b2 = *(const v2f*)(bcol + 2 * 16 * KS_STRIDE + kk * 4);
            v2f b3 = *(const v2f*)(bcol + 3 * 16 * KS_STRIDE + kk * 4);
            acc0 = __builtin_amdgcn_wmma_f32_16x16x4_f32(false, a[kk], false, b0, (short)0, acc0, false, false);
            acc1 = __builtin_amdgcn_wmma_f32_16x16x4_f32(false, a[kk], false, b1, (short)0, acc1, false, false);
            acc2 = __builtin_amdgcn_wmma_f32_16x16x4_f32(false, a[kk], false, b2, (short)0, acc2, false, false);
            acc3 = __builtin_amdgcn_wmma_f32_16x16x4_f32(false, a[kk], false, b3, (short)0, acc3, false, false);
        }
        __syncthreads();
    }

    float* outc = U + col0 + l16;
    const int rowbase = m0 + 8 * half;
#pragma unroll
    for (int v = 0; v < 8; ++v) {
        float* o = outc + (rowbase + v) * DSZ;
        o[0]  = acc0[v];
        o[16] = acc1[v];
        o[32] = acc2[v];
        o[48] = acc3[v];
    }
}

// ---------------------------------------------------------------------------
extern "C" void kernel_launch(void* const* d_in, const int* in_sizes, int n_in,
                              void* d_out, int out_size, void* d_ws, size_t ws_size,
                              hipStream_t stream) {
    const float* X      = (const float*)d_in[0];   // [512, 4096] fp32
    const float* thetas = (const float*)d_in[1];   // [130816] fp32
    float* U  = (float*)d_out;                     // [512, 4096] fp32

    float* cs = (float*)d_ws;                      // 130816 floats
    float* sn = cs + N_ROT;                        // 130816 floats
    float* Rm = sn + N_ROT;                        // 512*512 floats (1 MB)

    rot_cossin<<<(N_ROT + 255) / 256, 256, 0, stream>>>(thetas, cs, sn);

    rot_build_R<<<NSZ / C_TILE, 256, NSZ * C_TILE * sizeof(float), stream>>>(cs, sn, Rm);

    // 4 row-groups (of 8x16 rows) * 64 col-strips = 256 blocks, 8 waves each
    rot_wmma_gemm<<<256, 256, 0, stream>>>(Rm, X, U);
}